// NSE_layer_34969623724701
// MI455X (gfx1250) — compile-verified
//
#include <hip/hip_runtime.h>
#include <stdint.h>

// ---------------------------------------------------------------------------
// MI455X (gfx1250) NSE residual stencil.
// Memory-bound (~370 MB traffic -> ~16us @ 23.3 TB/s). No matrix structure,
// so the CDNA5-specific paths used are:
//   * async global->LDS DMA (global_load_async_to_lds_b128/b64, ASYNCcnt,
//     s_wait_asynccnt) to stage halo tiles into the 320KB-per-WGP LDS
//   * non-temporal (TH=NT) stores for the ~302MB write-only output streams,
//     keeping the 192MB L2 for the re-read lay/flow inputs.
// ---------------------------------------------------------------------------

#define NXc 2048
#define NYc 2048
#define NI  2046          // interior extent
#define TILE 32
#define LSTRIDE 36        // padded LDS row stride (multiple of 4 -> 16B row align)
#define LROWS 34
#define SEG (LROWS * LSTRIDE)   // 1224 dwords per array

// ---- CDNA5 async global->LDS (per-lane 16B / 8B DMA, tracked by ASYNCcnt) ----
__device__ __forceinline__ void async_b128(uint32_t lds, uint64_t g) {
  asm volatile("global_load_async_to_lds_b128 %0, %1, off"
               :: "v"(lds), "v"(g) : "memory");
}
__device__ __forceinline__ void async_b64(uint32_t lds, uint64_t g) {
  asm volatile("global_load_async_to_lds_b64 %0, %1, off"
               :: "v"(lds), "v"(g) : "memory");
}
__device__ __forceinline__ void wait_async0() {
  asm volatile("s_wait_asynccnt 0x0" ::: "memory");
}

__device__ __forceinline__ void st_nt(float* p, float v) {
  __builtin_nontemporal_store(v, p);
}

// ---------------------------------------------------------------------------
// Kernel A: row_in[y] = any(lay[y,:]==1), row_out[y] = any(lay[y,:]==3)
// ---------------------------------------------------------------------------
__global__ void row_flags_kernel(const int* __restrict__ lay,
                                 int* __restrict__ rin, int* __restrict__ rout) {
  __shared__ int s[2];
  const int y = blockIdx.x;
  if (threadIdx.x == 0) { s[0] = 0; s[1] = 0; }
  __syncthreads();
  const int* row = lay + (size_t)y * NXc;
  int a = 0, b = 0;
  for (int x = threadIdx.x; x < NXc; x += blockDim.x) {
    int l = row[x];
    a |= (l == 1);
    b |= (l == 3);
  }
  if (a) atomicOr(&s[0], 1);
  if (b) atomicOr(&s[1], 1);
  __syncthreads();
  if (threadIdx.x == 0) { rin[y] = s[0]; rout[y] = s[1]; }
}

// ---------------------------------------------------------------------------
// Kernel B: loss_nse (3,3,2046,2046). One 32x32 output tile per block.
// LDS holds lay + masked u,v,p on a 34x36 padded tile.
// ---------------------------------------------------------------------------
__global__ __launch_bounds__(256) void nse_loss_kernel(
    const int* __restrict__ lay, const float* __restrict__ flow,
    const int* __restrict__ rin, const int* __restrict__ rout,
    float* __restrict__ loss) {
  __shared__ __align__(16) int smem[4 * SEG];

  const int tid = threadIdx.x;
  const int bx = blockIdx.x, by = blockIdx.y;
  const int x0 = 1 + TILE * bx;           // first output column (global)
  const int y0 = 1 + TILE * by;           // first output row (global)
  const bool edge  = (bx < 63);           // need cols x0+31, x0+32 (b64 tail)
  const int nrows  = (by < 63) ? 34 : 32; // halo rows that exist in-bounds
  const int ncols  = edge ? 34 : 32;

  const uint32_t lbase = (uint32_t)(uintptr_t)smem; // low 32 bits = LDS offset
  const uint64_t layg  = (uint64_t)(uintptr_t)lay;
  const uint64_t flowg = (uint64_t)(uintptr_t)flow;
  const size_t P = (size_t)NXc * NYc;

  // ---- Phase 1: async DMA of lay,f0,f1,f2 halo tiles into LDS -------------
  // 32 aligned columns per row as 8x b128 (x0-1 = 32*bx -> 128B aligned),
  // plus an aligned b64 tail for cols x0+31..x0+32 when in-bounds.
  {
    const int chunk = tid & 7;
    for (int s = tid >> 3; s < 4 * nrows; s += 32) {
      const int arr = s & 3;
      const int r   = s >> 2;
      const int gy  = y0 - 1 + r;
      const uint64_t g = (arr == 0)
          ? layg  + 4ull * ((size_t)gy * NXc + (x0 - 1) + 4 * chunk)
          : flowg + 4ull * ((size_t)(arr - 1) * P + (size_t)gy * NXc + (x0 - 1) + 4 * chunk);
      const uint32_t l = lbase + 4u * (uint32_t)(arr * SEG + r * LSTRIDE + 4 * chunk);
      async_b128(l, g);
    }
  }
  if (edge) {
    for (int s = tid; s < 4 * nrows; s += 256) {
      const int arr = s & 3;
      const int r   = s >> 2;
      const int gy  = y0 - 1 + r;
      const uint64_t g = (arr == 0)
          ? layg  + 4ull * ((size_t)gy * NXc + (x0 + 31))
          : flowg + 4ull * ((size_t)(arr - 1) * P + (size_t)gy * NXc + (x0 + 31));
      const uint32_t l = lbase + 4u * (uint32_t)(arr * SEG + r * LSTRIDE + 32);
      async_b64(l, g);
    }
  }
  wait_async0();       // drain this wave's ASYNCcnt
  __syncthreads();     // make all waves' DMA visible

  int*   LAY = smem;
  float* U   = (float*)(smem + 1 * SEG);
  float* V   = (float*)(smem + 2 * SEG);
  float* Pp  = (float*)(smem + 3 * SEG);

  // ---- Phase 2: apply boundary masking in-place: x = flow*G_bc + G_bc_value
  {
    const int nel = nrows * ncols;
    for (int e = tid; e < 3 * nel; e += 256) {
      const int arr = e / nel;
      const int t   = e - arr * nel;
      const int r   = t / ncols;
      const int j   = t - r * ncols;
      const int gy  = y0 - 1 + r;
      const int gx  = x0 - 1 + j;
      const int li  = r * LSTRIDE + j;
      const int la  = LAY[li];
      float g = (la == 2) ? 0.f : 1.f;
      if (arr == 0) {
        float add = 0.f;
        if (gx == 0 && rin[gy]) { g = 0.f; add = 3.f; }
        U[li] = U[li] * g + add;
      } else if (arr == 1) {
        if (gx == 0 && rin[gy]) g = 0.f;
        V[li] = V[li] * g;
      } else {
        if (gx == NXc - 1 && rout[gy]) g = 0.f;
        Pp[li] = Pp[li] * g;
      }
    }
  }
  __syncthreads();

  // ---- Phase 3: stencil + per-case pressure gradient, 9 NT output stores --
  constexpr double Hd = 0.1 / 2047.0;
  const float inv_h = (float)(1.0 / Hd);
  const float nuh2  = (float)(0.05 / (Hd * Hd));
  const int lx  = tid & 31;
  const int lyb = tid >> 5;
  const size_t LP = (size_t)NI * NI;

  #pragma unroll
  for (int kk = 0; kk < 4; ++kk) {
    const int ly = lyb + 8 * kk;
    const int gx = x0 + lx, gy = y0 + ly;
    if (gx > NI || gy > NI) continue;
    const int r = ly + 1, j = lx + 1;
    const int li = r * LSTRIDE + j;
    const int la = LAY[li];
    float m0 = 0.f, m1 = 0.f, m2 = 0.f;
    if (la != 2) {  // lay==2 -> weight 0 -> all-zero outputs
      const float uc = U[li], un = U[li - LSTRIDE], us = U[li + LSTRIDE];
      const float uw = U[li - 1], ue = U[li + 1];
      const float vc = V[li], vn = V[li - LSTRIDE], vs = V[li + LSTRIDE];
      const float vw = V[li - 1], ve = V[li + 1];
      const float pc = Pp[li], pn = Pp[li - LSTRIDE], ps = Pp[li + LSTRIDE];
      const float pw = Pp[li - 1], pe = Pp[li + 1];
      const float dxu = 0.5f * (ue - uw), dyu = 0.5f * (us - un);
      const float dxv = 0.5f * (ve - vw), dyv = 0.5f * (vs - vn);
      const float lapu = un + us + uw + ue - 4.f * uc;
      const float lapv = vn + vs + vw + ve - 4.f * vc;
      float dpx, dpy;
      if (la == 4 || la == 8 || la == 11)       dpx = pe - pc;        // FDx
      else if (la == 6 || la == 9 || la == 10)  dpx = pc - pw;        // BDx
      else                                      dpx = 0.5f * (pe - pw);
      if (la == 7 || la == 10 || la == 11)      dpy = pn - pc;        // FDy
      else if (la == 5 || la == 8 || la == 9)   dpy = pc - ps;        // BDy
      else                                      dpy = 0.5f * (ps - pn);
      m0 = (uc * dxu + vc * dyu + dpx) * inv_h - nuh2 * lapu;
      m1 = (uc * dxv + vc * dyv + dpy) * inv_h - nuh2 * lapv;
      m2 = (dxu + dyv) * inv_h;
    }
    // loss_nse[i][c][gy-1][gx-1]; all 3 c-slices identical for interior pts
    float* o = loss + (size_t)(gy - 1) * NI + (gx - 1);
    st_nt(o,          m0); st_nt(o + 1*LP, m0); st_nt(o + 2*LP, m0);
    st_nt(o + 3*LP,   m1); st_nt(o + 4*LP, m1); st_nt(o + 5*LP, m1);
    st_nt(o + 6*LP,   m2); st_nt(o + 7*LP, m2); st_nt(o + 8*LP, m2);
  }
}

// ---------------------------------------------------------------------------
// Kernel C: G_bc_mask (1,3,NY,NX) and G_bc_v = stack([G_bc, G_bc_value])
// ---------------------------------------------------------------------------
__global__ void bc_outputs_kernel(const int* __restrict__ lay,
                                  const int* __restrict__ rin,
                                  const int* __restrict__ rout,
                                  float* __restrict__ mask_out,
                                  float* __restrict__ gbc_out,
                                  float* __restrict__ gval_out) {
  const int x = blockIdx.x * blockDim.x + threadIdx.x;
  const int y = blockIdx.y;
  const size_t idx = (size_t)y * NXc + x;
  const int la = lay[idx];
  const float base = (la == 2) ? 0.f : 1.f;
  float g0 = base, g1 = base, g2 = base, v0 = 0.f;
  if (x == 0 && rin[y])        { g0 = 0.f; g1 = 0.f; v0 = 3.f; }
  if (x == NXc - 1 && rout[y]) { g2 = 0.f; }
  const float code = (la > 3) ? (float)la : 0.f;
  const float z    = (la > 3) ? 0.f : 1.f;   // G_bc0 = G_bc * z
  const size_t P = (size_t)NXc * NYc;
  st_nt(mask_out + idx,         g0 * z + code);
  st_nt(mask_out + idx + P,     g1 * z + code);
  st_nt(mask_out + idx + 2 * P, g2 * z + code);
  st_nt(gbc_out + idx,          g0);
  st_nt(gbc_out + idx + P,      g1);
  st_nt(gbc_out + idx + 2 * P,  g2);
  st_nt(gval_out + idx,         v0);
  st_nt(gval_out + idx + P,     0.f);
  st_nt(gval_out + idx + 2 * P, 0.f);
}

// ---------------------------------------------------------------------------
extern "C" void kernel_launch(void* const* d_in, const int* in_sizes, int n_in,
                              void* d_out, int out_size, void* d_ws, size_t ws_size,
                              hipStream_t stream) {
  const int*   layout = (const int*)d_in[0];          // (1,2,NY,NX) int32
  const float* flow   = (const float*)d_in[1];        // (1,3,NY,NX) f32
  const int*   lay    = layout + (size_t)NXc * NYc;   // layout[0,1]

  int* rin  = (int*)d_ws;       // 2048 ints
  int* rout = rin + NYc;        // 2048 ints

  float* out  = (float*)d_out;
  const size_t LP = (size_t)NI * NI;
  const size_t P  = (size_t)NXc * NYc;
  float* loss = out;                 // 9 * 2046^2
  float* mask = out + 9 * LP;        // 3 * 2048^2
  float* gbc  = mask + 3 * P;        // 3 * 2048^2  (G_bc_v[0])
  float* gval = gbc + 3 * P;         // 3 * 2048^2  (G_bc_v[1])

  row_flags_kernel<<<NYc, 256, 0, stream>>>(lay, rin, rout);

  dim3 gridB(64, 64);
  nse_loss_kernel<<<gridB, 256, 0, stream>>>(lay, flow, rin, rout, loss);

  dim3 gridC(NXc / 256, NYc);
  bc_outputs_kernel<<<gridC, 256, 0, stream>>>(lay, rin, rout, mask, gbc, gval);
}